// AttentionNetwork_8564164788839
// MI455X (gfx1250) — compile-verified
//
#include <hip/hip_runtime.h>
#include <math.h>

// Problem dims (fixed by reference)
#define BATCH 4096
#define SEQ   200
#define DIN   256
#define DOUT  128
#define SP    208            // SEQ padded to multiple of 16 for WMMA M-tiles
#define XSTR  260            // LDS row stride (dwords); 260 % 64 = 4 -> conflict-free column reads
#define NORMF 0.08838834764831845f   // 1/sqrt(128)

typedef float v2f __attribute__((ext_vector_type(2)));
typedef float v4f __attribute__((ext_vector_type(4)));
typedef float v8f __attribute__((ext_vector_type(8)));
typedef int   v4i __attribute__((vector_size(16)));
typedef __attribute__((address_space(1))) v4i gv4i;   // global int4
typedef __attribute__((address_space(3))) v4i lv4i;   // LDS int4

#if __has_builtin(__builtin_amdgcn_global_load_async_to_lds_b128)
#define HAVE_ASYNC_LDS 1
#else
#define HAVE_ASYNC_LDS 0
#endif

// One workgroup (256 threads = 8 wave32) per batch element b.
// Phase 1: stage x[b] (200x256 f32) into LDS (async global->LDS if available).
// Phase 2: qk = Wk @ Q[b]  (VALU, 128 MACs/thread).
// Phase 3: scores = xs @ qk via V_WMMA_F32_16X16X4_F32 (13 M-tiles x 64 K-tiles).
// Phase 4: softmax over 200 scores (block reductions), write atten.
// Phase 5: xa = atten^T @ xs (VALU, conflict-free LDS columns).
// Phase 6: out = xa @ Wv + NORMF*bv via WMMA (8 N-tiles x 64 K-tiles).
__global__ __launch_bounds__(256) void attn_fused_kernel(
    const float* __restrict__ x, const float* __restrict__ Q,
    const float* __restrict__ Wk, const float* __restrict__ bk,
    const float* __restrict__ Wv, const float* __restrict__ bv,
    float* __restrict__ out, float* __restrict__ attn_out)
{
    __shared__ __align__(16) float xs[SP * XSTR];  // 216,320 B
    __shared__ float qs[DOUT];
    __shared__ float qk[DIN];
    __shared__ float sc[SP];
    __shared__ float attn[SP];
    __shared__ float xa[DIN];
    __shared__ float red[256];

    const int tid  = threadIdx.x;
    const int lane = tid & 31;
    const int wave = tid >> 5;
    const int b    = blockIdx.x;
    (void)bk; // softmax is invariant to the constant Q·bk shift

    // ---- Phase 1: stage x[b] into LDS, 16B per lane, fully coalesced ----
    const float* xb = x + (size_t)b * (SEQ * DIN);
#if HAVE_ASYNC_LDS
    #pragma unroll 5
    for (int it = 0; it < (SEQ * DIN) / (4 * 256); ++it) {  // 50 iters
        int e   = (tid + it * 256) * 4;
        int row = e >> 8;
        int col = e & 255;
        __builtin_amdgcn_global_load_async_to_lds_b128(
            (gv4i*)(xb + e),
            (lv4i*)&xs[row * XSTR + col],
            0, 0);
    }
    asm volatile("s_wait_asynccnt 0" ::: "memory");
#else
    #pragma unroll 5
    for (int it = 0; it < (SEQ * DIN) / (4 * 256); ++it) {  // 50 iters
        int e   = (tid + it * 256) * 4;
        int row = e >> 8;
        int col = e & 255;
        v4f v = *(const v4f*)(xb + e);
        *(v4f*)(&xs[row * XSTR + col]) = v;
    }
#endif
    // zero-fill padded rows 200..207
    for (int i = tid; i < (SP - SEQ) * DIN; i += 256) {
        xs[(SEQ + i / DIN) * XSTR + (i & (DIN - 1))] = 0.0f;
    }
    if (tid < DOUT) qs[tid] = Q[(size_t)b * DOUT + tid];
    __syncthreads();

    // ---- Phase 2: qk[i] = sum_d Wk[i,d] * Q[b,d]  (thread i = row i of Wk) ----
    {
        const float* wr = Wk + (size_t)tid * DOUT;
        float acc = 0.0f;
        #pragma unroll 8
        for (int d = 0; d < DOUT; ++d) acc = fmaf(wr[d], qs[d], acc);
        qk[tid] = acc;
    }
    __syncthreads();

    // ---- Phase 3: scores = xs @ qk via fp32 WMMA (N=1 in a 16-wide tile) ----
    // A 16x4 layout: lanes 0-15 hold K0,K1; lanes 16-31 hold K2,K3 (ISA 7.12.2)
    // B 4x16: column 0 lives in lane 0 (K0,K1) and lane 16 (K2,K3); other lanes 0.
    // B values loaded unconditionally (LDS broadcast) and masked via select --
    // avoids per-iteration EXEC save/restore around the WMMA.
    {
        const int  koff = (lane & 16) ? 2 : 0;
        const bool col0 = (lane & 15) == 0;
        for (int mt = wave; mt < SP / 16; mt += 8) {  // 13 tiles over 8 waves
            const float* xrow = &xs[(mt * 16 + (lane & 15)) * XSTR + koff];
            v8f c = {};
            #pragma unroll 4
            for (int kt = 0; kt < DIN / 4; ++kt) {
                const int k = kt * 4 + koff;
                v2f a, bb;
                a.x = xrow[kt * 4];
                a.y = xrow[kt * 4 + 1];
                const float q0 = qk[k];
                const float q1 = qk[k + 1];
                bb.x = col0 ? q0 : 0.0f;
                bb.y = col0 ? q1 : 0.0f;
                c = __builtin_amdgcn_wmma_f32_16x16x4_f32(
                        false, a, false, bb, (short)0, c, false, false);
            }
            // D column N=0: rows m0..m0+7 in lane 0, rows m0+8..m0+15 in lane 16
            if (lane == 0) {
                #pragma unroll
                for (int v = 0; v < 8; ++v) sc[mt * 16 + v] = c[v];
            } else if (lane == 16) {
                #pragma unroll
                for (int v = 0; v < 8; ++v) sc[mt * 16 + 8 + v] = c[v];
            }
        }
    }
    __syncthreads();

    // ---- Phase 4: softmax over SEQ scores, scaled by NORMF ----
    float sval = (tid < SEQ) ? sc[tid] : -INFINITY;
    red[tid] = sval;
    __syncthreads();
    for (int off = 128; off > 0; off >>= 1) {
        if (tid < off) red[tid] = fmaxf(red[tid], red[tid + off]);
        __syncthreads();
    }
    const float mx = red[0];
    __syncthreads();
    const float ev = (tid < SEQ) ? expf(sval - mx) : 0.0f;
    red[tid] = ev;
    __syncthreads();
    for (int off = 128; off > 0; off >>= 1) {
        if (tid < off) red[tid] += red[tid + off];
        __syncthreads();
    }
    const float inv = NORMF / red[0];
    __syncthreads();
    if (tid < SP)  attn[tid] = (tid < SEQ) ? ev * inv : 0.0f;
    if (tid < SEQ) attn_out[(size_t)b * SEQ + tid] = ev * inv;
    __syncthreads();

    // ---- Phase 5: xa[i] = sum_s attn[s] * xs[s,i] (thread i = column i) ----
    {
        float acc = 0.0f;
        #pragma unroll 8
        for (int s = 0; s < SEQ; ++s)
            acc = fmaf(attn[s], xs[s * XSTR + tid], acc);
        xa[tid] = acc;
    }
    __syncthreads();

    // ---- Phase 6: out[b] = xa @ Wv + NORMF*bv via fp32 WMMA (M=1 padded) ----
    // A row M=0 lives in lane 0 (K0,K1) and lane 16 (K2,K3); other rows zero.
    // A values loaded unconditionally (LDS broadcast) and masked via select.
    {
        const int  nt   = wave;                  // 8 waves -> 8 column tiles of 16
        const int  nc   = nt * 16 + (lane & 15);
        const int  koff = (lane & 16) ? 2 : 0;
        const bool col0 = (lane & 15) == 0;
        v8f c = {};
        #pragma unroll 4
        for (int kt = 0; kt < DIN / 4; ++kt) {
            const int k = kt * 4 + koff;
            v2f a, bb;
            const float a0 = xa[k];
            const float a1 = xa[k + 1];
            a.x = col0 ? a0 : 0.0f;
            a.y = col0 ? a1 : 0.0f;
            bb.x = Wv[(size_t)k * DOUT + nc];
            bb.y = Wv[(size_t)(k + 1) * DOUT + nc];
            c = __builtin_amdgcn_wmma_f32_16x16x4_f32(
                    false, a, false, bb, (short)0, c, false, false);
        }
        // D row M=0 = VGPR0, lanes 0-15, N = lane
        if (lane < 16)
            out[(size_t)b * DOUT + nc] = c[0] + NORMF * bv[nc];
    }
}

extern "C" void kernel_launch(void* const* d_in, const int* in_sizes, int n_in,
                              void* d_out, int out_size, void* d_ws, size_t ws_size,
                              hipStream_t stream) {
    (void)in_sizes; (void)n_in; (void)out_size; (void)d_ws; (void)ws_size;
    const float* x  = (const float*)d_in[0];
    const float* Q  = (const float*)d_in[1];
    const float* Wk = (const float*)d_in[2];
    const float* bk = (const float*)d_in[3];
    const float* Wv = (const float*)d_in[4];
    const float* bv = (const float*)d_in[5];
    float* out      = (float*)d_out;                       // [B, DOUT]
    float* attn_out = out + (size_t)BATCH * DOUT;          // [B, 1, SEQ]
    attn_fused_kernel<<<BATCH, 256, 0, stream>>>(x, Q, Wk, bk, Wv, bv, out, attn_out);
}